// LuongAttention_44109314130179
// MI455X (gfx1250) — compile-verified
//
#include <hip/hip_runtime.h>

typedef float v2f __attribute__((ext_vector_type(2)));
typedef float v8f __attribute__((ext_vector_type(8)));

#define B_    32
#define S_    2048
#define H_    1024
#define TWOH  2048
#define THREEH 3072

// ---------------------------------------------------------------------------
// Kernel 1: u = Q @ W_mem^T   (M=32, N=2048, K=1024), fp32 WMMA 16x16x4.
// One wave per 16x16 C tile. A[m,k]=Q[m*H+k]; B[k,n]=W_mem[n*H+k] (NT gemm,
// so both fragments are contiguous float2 loads).
// ---------------------------------------------------------------------------
__global__ void luong_proj_u(const float* __restrict__ Q,
                             const float* __restrict__ Wm,
                             float* __restrict__ U) {
  const int wave = threadIdx.x >> 5;
  const int lane = threadIdx.x & 31;
  const int tile = blockIdx.x * 4 + wave;     // 0..255
  const int mt   = (tile & 1) * 16;           // 0 / 16
  const int nt   = (tile >> 1) * 16;          // 0..2032
  const int half = lane >> 4;                 // 0: K 0,1  1: K 2,3
  const int l15  = lane & 15;

  const float* aptr = Q  + (mt + l15) * H_ + half * 2;
  const float* bptr = Wm + (nt + l15) * H_ + half * 2;

  v8f acc = {0.f, 0.f, 0.f, 0.f, 0.f, 0.f, 0.f, 0.f};
  for (int k = 0; k < H_; k += 4) {
    v2f a = *(const v2f*)(aptr + k);
    v2f b = *(const v2f*)(bptr + k);
    acc = __builtin_amdgcn_wmma_f32_16x16x4_f32(
        false, a, false, b, (short)0, acc, false, false);
  }

  const int mrow = mt + half * 8;             // lanes 0-15: M=r, 16-31: M=8+r
  const int ncol = nt + l15;
#pragma unroll
  for (int r = 0; r < 8; ++r)
    U[(mrow + r) * TWOH + ncol] = acc[r];
}

// ---------------------------------------------------------------------------
// Kernel 2: scores[b,s] = E[b,s,:] . u[b,:] + mask*NEG_INF.
// One wave per (b,s); u row staged in LDS; float4 streaming of E.
// ---------------------------------------------------------------------------
__global__ void luong_scores(const float* __restrict__ E,
                             const float* __restrict__ U,
                             const float* __restrict__ Mask,
                             float* __restrict__ Sc) {
  __shared__ float us[TWOH];
  const int b = blockIdx.y;
  for (int i = threadIdx.x; i < TWOH; i += 256) us[i] = U[b * TWOH + i];
  __syncthreads();

  const int wave = threadIdx.x >> 5;
  const int lane = threadIdx.x & 31;
  const int s = blockIdx.x * 8 + wave;
  const float* ep = E + ((size_t)b * S_ + s) * TWOH;

  float acc = 0.f;
  for (int k = lane * 4; k < TWOH; k += 128) {
    float4 v = *(const float4*)(ep + k);
    acc += v.x * us[k] + v.y * us[k + 1] + v.z * us[k + 2] + v.w * us[k + 3];
  }
#pragma unroll
  for (int off = 16; off > 0; off >>= 1)
    acc += __shfl_xor(acc, off, 32);
  if (lane == 0)
    Sc[b * S_ + s] = acc + Mask[b * S_ + s] * -1e9f;
}

// ---------------------------------------------------------------------------
// Kernel 3: softmax over S per batch; weights written straight into d_out.
// ---------------------------------------------------------------------------
__global__ void luong_softmax(const float* __restrict__ Sc,
                              float* __restrict__ W) {
  __shared__ float red[256];
  const int b = blockIdx.x, tid = threadIdx.x;

  float v[8];
  float mx = -3.4e38f;
#pragma unroll
  for (int i = 0; i < 8; ++i) {
    v[i] = Sc[b * S_ + tid + i * 256];
    mx = fmaxf(mx, v[i]);
  }
  red[tid] = mx; __syncthreads();
  for (int off = 128; off > 0; off >>= 1) {
    if (tid < off) red[tid] = fmaxf(red[tid], red[tid + off]);
    __syncthreads();
  }
  mx = red[0]; __syncthreads();

  float sum = 0.f;
#pragma unroll
  for (int i = 0; i < 8; ++i) { v[i] = __expf(v[i] - mx); sum += v[i]; }
  red[tid] = sum; __syncthreads();
  for (int off = 128; off > 0; off >>= 1) {
    if (tid < off) red[tid] += red[tid + off];
    __syncthreads();
  }
  const float inv = 1.f / red[0];
#pragma unroll
  for (int i = 0; i < 8; ++i)
    W[b * S_ + tid + i * 256] = v[i] * inv;
}

// ---------------------------------------------------------------------------
// Kernel 4: context[b,k] = sum_s w[b,s] * E[b,s,k]. Second streaming pass.
// Weights staged in LDS; E loads coalesced across k.
// ---------------------------------------------------------------------------
__global__ void luong_context(const float* __restrict__ E,
                              const float* __restrict__ W,
                              float* __restrict__ Ctx) {
  __shared__ float wsh[S_];
  const int b = blockIdx.y;
  for (int i = threadIdx.x; i < S_; i += 256) wsh[i] = W[b * S_ + i];
  __syncthreads();

  const int k = blockIdx.x * 256 + threadIdx.x;
  const float* ep = E + (size_t)b * S_ * TWOH + k;
  float acc = 0.f;
  for (int s = 0; s < S_; ++s)
    acc += wsh[s] * ep[(size_t)s * TWOH];
  Ctx[b * TWOH + k] = acc;
}

// ---------------------------------------------------------------------------
// Kernel 5: pack cat = [query ; context]  ([32,3072])
// ---------------------------------------------------------------------------
__global__ void luong_pack(const float* __restrict__ Q,
                           const float* __restrict__ Ctx,
                           float* __restrict__ Cat) {
  const int idx = blockIdx.x * 256 + threadIdx.x;  // < 32*3072
  const int m = idx / THREEH, h = idx % THREEH;
  Cat[idx] = (h < H_) ? Q[m * H_ + h] : Ctx[m * TWOH + (h - H_)];
}

// ---------------------------------------------------------------------------
// Kernel 6: out = cat @ W_att  (M=32, N=1024, K=3072), fp32 WMMA 16x16x4.
// A[m,k]=cat[m*3072+k]; B[k,n]=W_att[k*1024+n] (NN gemm).
// ---------------------------------------------------------------------------
__global__ void luong_outproj(const float* __restrict__ Cat,
                              const float* __restrict__ Wa,
                              float* __restrict__ Out) {
  const int wave = threadIdx.x >> 5;
  const int lane = threadIdx.x & 31;
  const int tile = blockIdx.x * 4 + wave;     // 0..127
  const int mt   = (tile & 1) * 16;
  const int nt   = (tile >> 1) * 16;          // 0..1008
  const int half = lane >> 4;
  const int l15  = lane & 15;

  const float* aptr = Cat + (mt + l15) * THREEH + half * 2;
  const float* bptr = Wa + nt + l15;

  v8f acc = {0.f, 0.f, 0.f, 0.f, 0.f, 0.f, 0.f, 0.f};
  for (int k = 0; k < THREEH; k += 4) {
    v2f a = *(const v2f*)(aptr + k);
    const int kb = k + half * 2;
    v2f b;
    b.x = bptr[(size_t)kb * H_];
    b.y = bptr[(size_t)(kb + 1) * H_];
    acc = __builtin_amdgcn_wmma_f32_16x16x4_f32(
        false, a, false, b, (short)0, acc, false, false);
  }

  const int mrow = mt + half * 8;
  const int ncol = nt + l15;
#pragma unroll
  for (int r = 0; r < 8; ++r)
    Out[(mrow + r) * H_ + ncol] = acc[r];
}

// ---------------------------------------------------------------------------
extern "C" void kernel_launch(void* const* d_in, const int* in_sizes, int n_in,
                              void* d_out, int out_size, void* d_ws, size_t ws_size,
                              hipStream_t stream) {
  (void)in_sizes; (void)n_in; (void)out_size; (void)ws_size;

  const float* q   = (const float*)d_in[0];   // [32,1024]
  const float* e   = (const float*)d_in[1];   // [32,2048,2048]
  const float* msk = (const float*)d_in[2];   // [32,2048]
  const float* wm  = (const float*)d_in[3];   // [2048,1024]
  const float* wa  = (const float*)d_in[4];   // [3072,1024]

  float* out  = (float*)d_out;                // attention_outputs [32,1024]
  float* outw = out + B_ * H_;                // attention_weights [32,2048]

  float* ws  = (float*)d_ws;
  float* u   = ws;                            // 65536 floats
  float* sc  = ws + 65536;                    // 65536 floats
  float* ctx = ws + 131072;                   // 65536 floats
  float* cat = ws + 196608;                   // 98304 floats

  luong_proj_u <<<dim3(64),            dim3(128), 0, stream>>>(q, wm, u);
  luong_scores <<<dim3(S_ / 8, B_),    dim3(256), 0, stream>>>(e, u, msk, sc);
  luong_softmax<<<dim3(B_),            dim3(256), 0, stream>>>(sc, outw);
  luong_context<<<dim3(TWOH / 256, B_),dim3(256), 0, stream>>>(e, outw, ctx);
  luong_pack   <<<dim3(B_ * THREEH / 256), dim3(256), 0, stream>>>(q, ctx, cat);
  luong_outproj<<<dim3(32),            dim3(128), 0, stream>>>(cat, wa, out);
}